// MessagePassing_69097433858446
// MI455X (gfx1250) — compile-verified
//
#include <hip/hip_runtime.h>
#include <cstdint>
#include <cstddef>

#define D_FEAT 64
#define TILE_E 128           // edges staged per block (32 KB of edge_attr in LDS)
#define BLOCK  256           // 8 wave32 waves
#define CPOL_NT 1            // CPol TH=NT: non-temporal streaming load hint

// ---- CDNA5 async global->LDS path (guarded so both toolchains compile) ----
#if defined(__HIP_DEVICE_COMPILE__) && __has_builtin(__builtin_amdgcn_global_load_async_to_lds_b128)
#define HAVE_ASYNC 1
typedef int v4i __attribute__((ext_vector_type(4)));
typedef __attribute__((address_space(1))) v4i* g4_ptr;   // global int4*
typedef __attribute__((address_space(3))) v4i* l4_ptr;   // LDS int4*
#else
#define HAVE_ASYNC 0
#endif

// Hardware no-return f32 atomic add into global memory.
__device__ __forceinline__ void atomic_add_f32(float* p, float v) {
#if defined(__HIP_DEVICE_COMPILE__) && __has_builtin(__builtin_amdgcn_global_atomic_fadd_f32)
  (void)__builtin_amdgcn_global_atomic_fadd_f32(
      (__attribute__((address_space(1))) float*)p, v);
#else
  (void)unsafeAtomicAdd(p, v);
#endif
}

__global__ void zero_f32(float* __restrict__ out, int n) {
  int i = (blockIdx.x * blockDim.x + threadIdx.x) * 4;
  if (i + 3 < n) {
    *(float4*)(out + i) = make_float4(0.f, 0.f, 0.f, 0.f);
  } else {
    for (int k = i; k < n; ++k) out[k] = 0.f;
  }
}

// One 128-edge tile: async-stage edge_attr, stage indices, then
// message + scatter-add. FULL=true -> no per-lane bounds checks.
template <bool FULL>
__device__ __forceinline__ void scatter_tile(
    const float* __restrict__ x,
    const float* __restrict__ ea,
    const long long* __restrict__ eidx,
    float* __restrict__ out,
    int E, int e0, int t,
    int* s_src, int* s_dst, float* s_attr)
{
#if HAVE_ASYNC
  // Phase A1: async-copy 32 KB edge_attr tile into LDS.
  // Thread t owns 128 contiguous bytes (half an edge row) => 8 x b128 async ops.
  if (FULL || (e0 + (t >> 1)) < E) {
    g4_ptr gp = (g4_ptr)(ea + (size_t)e0 * D_FEAT + (size_t)t * 32);
    l4_ptr lp = (l4_ptr)(&s_attr[t * 32]);
    __builtin_amdgcn_global_load_async_to_lds_b128(gp, lp,   0, CPOL_NT);
    __builtin_amdgcn_global_load_async_to_lds_b128(gp, lp,  16, CPOL_NT);
    __builtin_amdgcn_global_load_async_to_lds_b128(gp, lp,  32, CPOL_NT);
    __builtin_amdgcn_global_load_async_to_lds_b128(gp, lp,  48, CPOL_NT);
    __builtin_amdgcn_global_load_async_to_lds_b128(gp, lp,  64, CPOL_NT);
    __builtin_amdgcn_global_load_async_to_lds_b128(gp, lp,  80, CPOL_NT);
    __builtin_amdgcn_global_load_async_to_lds_b128(gp, lp,  96, CPOL_NT);
    __builtin_amdgcn_global_load_async_to_lds_b128(gp, lp, 112, CPOL_NT);
  }
#endif

  // Phase A2 (overlaps with async copies): stage int64 edge indices in LDS.
  {
    const int i     = t & (TILE_E - 1);
    const int which = t >> 7;                 // 0 = src row, 1 = dst row
    long long v = 0;
    if (FULL || (e0 + i) < E)
      v = eidx[(size_t)which * (size_t)E + (size_t)(e0 + i)];
    if (which == 0) s_src[i] = (int)v; else s_dst[i] = (int)v;
  }

#if HAVE_ASYNC
#if __has_builtin(__builtin_amdgcn_s_wait_asynccnt)
  __builtin_amdgcn_s_wait_asynccnt(0);
#else
  asm volatile("s_wait_asynccnt 0x0" ::: "memory");
#endif
#endif
  __syncthreads();

  // Phase B: 128 edges x 16 float4-quads = 2048 tasks over 256 threads.
  int task = t;
#pragma unroll
  for (int it = 0; it < (TILE_E * (D_FEAT / 4)) / BLOCK; ++it, task += BLOCK) {
    const int el = task >> 4;                 // local edge 0..127
    const int q4 = (task & 15) << 2;          // feature offset 0,4,...,60
    const int e  = e0 + el;
    if (FULL || e < E) {
      const int src = s_src[el];
      const int dst = s_dst[el];
#if HAVE_ASYNC
      const float4 a = *(const float4*)&s_attr[el * D_FEAT + q4];
#else
      const float4 a = *(const float4*)&ea[(size_t)e * D_FEAT + q4];
#endif
      const float4 xs = *(const float4*)&x[(size_t)src * D_FEAT + q4];  // L2-resident gather
      const float4 xd = *(const float4*)&x[(size_t)dst * D_FEAT + q4];  // L2-resident gather
      float* o = out + (size_t)dst * D_FEAT + q4;
      atomic_add_f32(o + 0, a.x + xs.x + xd.x);
      atomic_add_f32(o + 1, a.y + xs.y + xd.y);
      atomic_add_f32(o + 2, a.z + xs.z + xd.z);
      atomic_add_f32(o + 3, a.w + xs.w + xd.w);
    }
  }
}

// out[dst] += x[src] + x[dst] + edge_attr[e], scatter-add over 1.25M edges.
__global__ void __launch_bounds__(BLOCK) gnn_scatter(
    const float* __restrict__ x,
    const float* __restrict__ ea,
    const long long* __restrict__ eidx,   // [2, E] int64: row 0 = src, row 1 = dst
    float* __restrict__ out,
    int E)
{
  __shared__ int s_src[TILE_E];
  __shared__ int s_dst[TILE_E];
#if HAVE_ASYNC
  __shared__ float s_attr[TILE_E * D_FEAT];   // 32 KB staged edge_attr tile
  float* s_attr_p = s_attr;
#else
  float* s_attr_p = nullptr;
#endif

  const int t  = threadIdx.x;
  const int e0 = blockIdx.x * TILE_E;

  if (e0 + TILE_E <= E) {
    // Steady state (9765/9766 blocks): uniform branch, zero per-lane guards.
    scatter_tile<true>(x, ea, eidx, out, E, e0, t, s_src, s_dst, s_attr_p);
  } else {
    scatter_tile<false>(x, ea, eidx, out, E, e0, t, s_src, s_dst, s_attr_p);
  }
}

extern "C" void kernel_launch(void* const* d_in, const int* in_sizes, int n_in,
                              void* d_out, int out_size, void* d_ws, size_t ws_size,
                              hipStream_t stream) {
  const float*     x    = (const float*)d_in[0];      // [N, 64] f32
  const float*     ea   = (const float*)d_in[1];      // [E, 64] f32
  const long long* eidx = (const long long*)d_in[2];  // [2, E] i64
  float*           out  = (float*)d_out;              // [N, 64] f32

  const int E = in_sizes[2] / 2;

  // Output accumulator must start at zero (harness poisons d_out).
  {
    const int vec    = (out_size + 3) / 4;
    const int blocks = (vec + 255) / 256;
    zero_f32<<<blocks, 256, 0, stream>>>(out, out_size);
  }

  const int blocks = (E + TILE_E - 1) / TILE_E;
  gnn_scatter<<<blocks, BLOCK, 0, stream>>>(x, ea, eidx, out, E);
}